// _LocalBatchTopKManifoldSAE_18966575579195
// MI455X (gfx1250) — compile-verified
//
#include <hip/hip_runtime.h>
#include <hip/hip_bf16.h>

typedef float v2f __attribute__((ext_vector_type(2)));
typedef float v8f __attribute__((ext_vector_type(8)));
typedef unsigned int u32x4 __attribute__((ext_vector_type(4)));
typedef int i32x4 __attribute__((ext_vector_type(4)));
typedef int i32x8 __attribute__((ext_vector_type(8)));

#define D_MODEL   2048
#define N_GROUPS  2048
#define GROUP_R   8
#define LATENT    16384
#define N_TOKENS  8192
#define K_GROUPS  64
#define CHUNK     512
#define EPS       1e-6f

#if defined(__has_builtin)
#  if __has_builtin(__builtin_amdgcn_tensor_load_to_lds)
#    define HAS_TDM 1
#  else
#    define HAS_TDM 0
#  endif
#else
#  define HAS_TDM 0
#endif

// ---------------------------------------------------------------------------
// Kernel 1: per-row layernorm-style normalize. One block (256 thr) per token.
// ---------------------------------------------------------------------------
__global__ __launch_bounds__(256) void ln_kernel(const float* __restrict__ x,
                                                 float* __restrict__ y,
                                                 float* __restrict__ mu_out,
                                                 float* __restrict__ std_out) {
    const int tok = blockIdx.x;
    const int tid = threadIdx.x;
    const float* row = x + (size_t)tok * D_MODEL;

    float4 a = *(const float4*)(row + tid * 4);
    float4 b = *(const float4*)(row + 1024 + tid * 4);

    float s1 = a.x + a.y + a.z + a.w + b.x + b.y + b.z + b.w;
    float s2 = a.x * a.x + a.y * a.y + a.z * a.z + a.w * a.w +
               b.x * b.x + b.y * b.y + b.z * b.z + b.w * b.w;

    __shared__ float r1[256];
    __shared__ float r2[256];
    r1[tid] = s1; r2[tid] = s2;
    __syncthreads();
    for (int s = 128; s > 0; s >>= 1) {
        if (tid < s) { r1[tid] += r1[tid + s]; r2[tid] += r2[tid + s]; }
        __syncthreads();
    }
    const float mu  = r1[0] * (1.0f / D_MODEL);
    float var = r2[0] * (1.0f / D_MODEL) - mu * mu;
    var = var > 0.0f ? var : 0.0f;
    const float sd  = fmaxf(sqrtf(var), EPS);
    const float inv = 1.0f / sd;

    float* yrow = y + (size_t)tok * D_MODEL;
    float4 oa = make_float4((a.x - mu) * inv, (a.y - mu) * inv, (a.z - mu) * inv, (a.w - mu) * inv);
    float4 ob = make_float4((b.x - mu) * inv, (b.y - mu) * inv, (b.z - mu) * inv, (b.w - mu) * inv);
    *(float4*)(yrow + tid * 4) = oa;
    *(float4*)(yrow + 1024 + tid * 4) = ob;
    if (tid == 0) { mu_out[tok] = mu; std_out[tok] = sd; }
}

// ---------------------------------------------------------------------------
// TDM: issue a 2-D tile load  global[tile_h x tile_w, row stride ld elems]
//      -> LDS at byte offset lds_byte, rows padded via D# pad fields.
// Descriptor layout per CDNA5 ISA.
// ---------------------------------------------------------------------------
#if HAS_TDM
__device__ __forceinline__ void tdm_load_2d(unsigned lds_byte, const void* gptr,
                                            unsigned tile_w, unsigned tile_h,
                                            unsigned ld_elems,
                                            unsigned pad_int_code, unsigned pad_amt_code) {
    const unsigned long long ga = (unsigned long long)(uintptr_t)gptr;
    u32x4 g0;
    g0[0] = 1u;                                           // count=1 (valid), user mode
    g0[1] = lds_byte;                                     // lds_addr
    g0[2] = (unsigned)(ga & 0xFFFFFFFFu);                 // global_addr[31:0]
    g0[3] = (unsigned)((ga >> 32) & 0x01FFFFFFu) | (2u << 30);  // global_addr[56:32] | type=2

    i32x8 g1;
    g1[0] = (int)((2u << 16) |                            // data_size = 4 bytes
                  (1u << 20) |                            // pad_enable
                  (pad_int_code << 22) |                  // pad_interval code
                  (pad_amt_code << 25));                  // pad_amount code
    g1[1] = (int)((tile_w & 0xFFFFu) << 16);              // tensor_dim0[15:0] (atomic barrier addr = 0)
    g1[2] = (int)(((tile_w >> 16) & 0xFFFFu) |            // tensor_dim0[31:16]
                  ((tile_h & 0xFFFFu) << 16));            // tensor_dim1[15:0]
    g1[3] = (int)(((tile_h >> 16) & 0xFFFFu) |            // tensor_dim1[31:16]
                  ((tile_w & 0xFFFFu) << 16));            // tile_dim0
    g1[4] = (int)(tile_h & 0xFFFFu);                      // tile_dim1 (tile_dim2 = 0 -> unused)
    g1[5] = (int)ld_elems;                                // tensor_dim0_stride[31:0]
    g1[6] = 0;                                            // stride[47:32]=0, dim1 stride lo = 0
    g1[7] = 0;

    const i32x4 z4 = (i32x4){0, 0, 0, 0};
#if __clang_major__ >= 23
    const i32x8 z8 = (i32x8){0, 0, 0, 0, 0, 0, 0, 0};
    __builtin_amdgcn_tensor_load_to_lds(g0, g1, z4, z4, z8, 0);
#else
    __builtin_amdgcn_tensor_load_to_lds(g0, g1, z4, z4, 0);
#endif
}
#endif

// ---------------------------------------------------------------------------
// Kernel 2: tiled fp32 WMMA GEMM  C[M,N] = A[M,K] * B[K,N] + bias
// 256 thr (8 waves), tile 128x128, K-step 16, double-buffered TDM staging.
// Wave grid 4x2; each wave computes 32x64 = 2x4 tiles via V_WMMA_F32_16X16X4_F32.
// epi != 0: C = (acc + bias[n]) * stdv[m] + mu[m]
// ---------------------------------------------------------------------------
#define ASTR 20    // 16 dwords + pad 4  (TDM: pad_interval=3 -> 16 dw, pad_amount=3 -> 4 dw)
#define BSTR 132   // 128 dwords + pad 4 (TDM: pad_interval=6 -> 128 dw, pad_amount=3 -> 4 dw)

__global__ __launch_bounds__(256) void gemm_wmma_kernel(
    const float* __restrict__ A, int lda,
    const float* __restrict__ B, int ldb,
    const float* __restrict__ bias,
    float* __restrict__ C, int ldc,
    int K,
    const float* __restrict__ stdv, const float* __restrict__ mu, int epi) {

    __shared__ float As[2][128 * ASTR];
    __shared__ float Bs[2][16 * BSTR];

    const int tid  = threadIdx.x;
    const int lane = tid & 31;
    const int w    = tid >> 5;
    const int wm   = w >> 1;          // 0..3 -> rows wm*32
    const int wn   = w & 1;           // 0..1 -> cols wn*64
    const int lm   = lane & 15;
    const int lh   = lane >> 4;       // selects K pair (A/B) and M half (C)

    const int n0 = blockIdx.x * 128;
    const int m0 = blockIdx.y * 128;

    v8f acc[2][4];
#pragma unroll
    for (int i = 0; i < 2; ++i)
#pragma unroll
        for (int j = 0; j < 4; ++j)
            acc[i][j] = (v8f){0.f,0.f,0.f,0.f,0.f,0.f,0.f,0.f};

    int cur = 0;
#if HAS_TDM
    // Prime buffer 0: A tile = 128 rows x 16 cols, B tile = 16 rows x 128 cols.
    if (w == 0) {
        tdm_load_2d((unsigned)(uintptr_t)&As[0][0], A + (size_t)m0 * lda, 16, 128,
                    (unsigned)lda, 3u, 3u);
        tdm_load_2d((unsigned)(uintptr_t)&Bs[0][0], B + n0, 128, 16,
                    (unsigned)ldb, 6u, 3u);
    }
#endif

    for (int kb = 0; kb < K; kb += 16) {
#if HAS_TDM
        if (w == 0) __builtin_amdgcn_s_wait_tensorcnt(0);
        __syncthreads();                       // current buffer visible to all waves
        if (w == 0 && kb + 16 < K) {           // overlap next DMA with compute
            const int nxt = cur ^ 1;
            tdm_load_2d((unsigned)(uintptr_t)&As[nxt][0],
                        A + (size_t)m0 * lda + kb + 16, 16, 128, (unsigned)lda, 3u, 3u);
            tdm_load_2d((unsigned)(uintptr_t)&Bs[nxt][0],
                        B + (size_t)(kb + 16) * ldb + n0, 128, 16, (unsigned)ldb, 6u, 3u);
        }
#else
        // Fallback cooperative staging (single buffer)
#pragma unroll
        for (int j = 0; j < 2; ++j) {
            int i   = tid + j * 256;
            int row = i >> 2;
            int seg = i & 3;
            float4 v = *(const float4*)(A + (size_t)(m0 + row) * lda + kb + seg * 4);
            *(float4*)&As[0][row * ASTR + seg * 4] = v;
            if (kb + 16 < K)
                __builtin_prefetch(A + (size_t)(m0 + row) * lda + kb + 16 + seg * 4, 0, 1);
        }
#pragma unroll
        for (int j = 0; j < 2; ++j) {
            int i    = tid + j * 256;
            int kk   = i >> 5;
            int nseg = i & 31;
            float4 v = *(const float4*)(B + (size_t)(kb + kk) * ldb + n0 + nseg * 4);
            *(float4*)&Bs[0][kk * BSTR + nseg * 4] = v;
            if (kb + 16 < K)
                __builtin_prefetch(B + (size_t)(kb + 16 + kk) * ldb + n0 + nseg * 4, 0, 1);
        }
        __syncthreads();
#endif

        const float* Asb = &As[cur][0];
        const float* Bsb = &Bs[cur][0];
#pragma unroll
        for (int ks = 0; ks < 16; ks += 4) {
            const int krow = ks + lh * 2;
            v2f afrag[2];
#pragma unroll
            for (int mt = 0; mt < 2; ++mt)
                afrag[mt] = *(const v2f*)&Asb[(wm * 32 + mt * 16 + lm) * ASTR + krow];
            v2f bfrag[4];
#pragma unroll
            for (int nt = 0; nt < 4; ++nt) {
                int col = wn * 64 + nt * 16 + lm;
                bfrag[nt].x = Bsb[krow * BSTR + col];
                bfrag[nt].y = Bsb[(krow + 1) * BSTR + col];
            }
#pragma unroll
            for (int mt = 0; mt < 2; ++mt)
#pragma unroll
                for (int nt = 0; nt < 4; ++nt)
                    acc[mt][nt] = __builtin_amdgcn_wmma_f32_16x16x4_f32(
                        false, afrag[mt], false, bfrag[nt],
                        (short)0, acc[mt][nt], false, false);
        }

#if HAS_TDM
        cur ^= 1;                              // readers done next iter after barrier
#else
        __syncthreads();
#endif
    }

    // ---- epilogue: bias (+ optional un-normalize) and store
#pragma unroll
    for (int mt = 0; mt < 2; ++mt) {
#pragma unroll
        for (int nt = 0; nt < 4; ++nt) {
            const int n  = n0 + wn * 64 + nt * 16 + lm;
            const float bv = bias[n];
#pragma unroll
            for (int j = 0; j < 8; ++j) {
                const int m = m0 + wm * 32 + mt * 16 + lh * 8 + j;
                float v = acc[mt][nt][j] + bv;
                if (epi) v = v * stdv[m] + mu[m];
                C[(size_t)m * ldc + n] = v;
            }
        }
    }
}

// ---------------------------------------------------------------------------
// Kernel 3: per-token group top-k mask (in place on pre_acts).
// Keys = group sum-of-squares (monotone in the norm); 32-step block-uniform
// binary search on the non-negative float bit pattern; zero losers in place.
// ---------------------------------------------------------------------------
__global__ __launch_bounds__(256) void topk_mask_kernel(float* __restrict__ pre) {
    const int tok = blockIdx.x;
    const int tid = threadIdx.x;
    float* row  = pre + (size_t)tok * LATENT;
    float* base = row + tid * 64;

    float4 v[16];
#pragma unroll
    for (int i = 0; i < 16; ++i) v[i] = *(const float4*)(base + i * 4);

    unsigned keys[8];
#pragma unroll
    for (int g = 0; g < 8; ++g) {
        float4 p = v[2 * g], q = v[2 * g + 1];
        float ss = p.x * p.x + p.y * p.y + p.z * p.z + p.w * p.w +
                   q.x * q.x + q.y * q.y + q.z * q.z + q.w * q.w;
        keys[g] = __float_as_uint(ss);
    }

    __shared__ unsigned cnt;
    unsigned lo = 0u, hi = 0xFFFFFFFFu;
    for (int it = 0; it < 32 && lo < hi; ++it) {
        unsigned mid = lo + (unsigned)((((unsigned long long)(hi - lo)) + 1ull) >> 1);
        __syncthreads();
        if (tid == 0) cnt = 0u;
        __syncthreads();
        unsigned loc = 0u;
#pragma unroll
        for (int g = 0; g < 8; ++g) loc += (keys[g] >= mid) ? 1u : 0u;
        if (loc) atomicAdd(&cnt, loc);
        __syncthreads();
        unsigned c = cnt;
        if (c >= (unsigned)K_GROUPS) lo = mid; else hi = mid - 1u;
    }
    const unsigned thr = lo;

    const float4 z = make_float4(0.f, 0.f, 0.f, 0.f);
#pragma unroll
    for (int g = 0; g < 8; ++g) {
        if (keys[g] < thr) {
            *(float4*)(base + g * 8)     = z;
            *(float4*)(base + g * 8 + 4) = z;
        }
    }
}

// ---------------------------------------------------------------------------
// Host-side launch
// ---------------------------------------------------------------------------
extern "C" void kernel_launch(void* const* d_in, const int* in_sizes, int n_in,
                              void* d_out, int out_size, void* d_ws, size_t ws_size,
                              hipStream_t stream) {
    const float* x     = (const float*)d_in[0];
    const float* W_enc = (const float*)d_in[1];
    const float* W_dec = (const float*)d_in[2];
    const float* b_enc = (const float*)d_in[3];
    const float* b_dec = (const float*)d_in[4];
    float* out = (float*)d_out;

    // workspace layout (floats): sae_in | mu | std | pre_chunk  (~101 MB)
    float* ws      = (float*)d_ws;
    float* sae_in  = ws;                                         // 8192*2048
    float* muv     = sae_in + (size_t)N_TOKENS * D_MODEL;        // 8192
    float* stv     = muv + N_TOKENS;                             // 8192
    float* pre     = stv + N_TOKENS;                             // CHUNK*16384

    ln_kernel<<<N_TOKENS, 256, 0, stream>>>(x, sae_in, muv, stv);

    for (int c = 0; c < N_TOKENS / CHUNK; ++c) {
        const int t0 = c * CHUNK;
        gemm_wmma_kernel<<<dim3(LATENT / 128, CHUNK / 128), 256, 0, stream>>>(
            sae_in + (size_t)t0 * D_MODEL, D_MODEL,
            W_enc, LATENT, b_enc,
            pre, LATENT, D_MODEL,
            nullptr, nullptr, 0);
        topk_mask_kernel<<<CHUNK, 256, 0, stream>>>(pre);
        gemm_wmma_kernel<<<dim3(D_MODEL / 128, CHUNK / 128), 256, 0, stream>>>(
            pre, LATENT,
            W_dec, D_MODEL, b_dec,
            out + (size_t)t0 * D_MODEL, D_MODEL, LATENT,
            stv + t0, muv + t0, 1);
    }
    (void)in_sizes; (void)n_in; (void)out_size; (void)ws_size;
}